// En_STGNN_40252433498167
// MI455X (gfx1250) — compile-verified
//
#include <hip/hip_runtime.h>
#include <math.h>

typedef __attribute__((ext_vector_type(2))) float v2f;
typedef __attribute__((ext_vector_type(8))) float v8f;

// ---- model constants (from reference setup) ----
constexpr int CB = 8, CC = 63, CF = 5, CS = 2048;
constexpr int CL1 = 1023, CL2 = 511, CL3 = 255;
constexpr int CE = 315, CHD = 105;             // embed dim, head dim (3 heads)
constexpr int CO = 32, CH = 64;
constexpr int CG = CB * CL1;                   // 8184 graphs
constexpr int CN = CG * CC;                    // 515592 nodes

// =====================================================================
// Generic f32 WMMA GEMM:  C[M,N] = A[M,K] * op(B) (+bias) (+relu)
//  - A row-major (lda), B row-major: TRANSB ? B[N,K] : B[K,N] (ldb)
//  - batch offsets: ptr += blockIdx.y*s?y + blockIdx.z*s?z
//  - one 16x32 tile per wave (2 accumulators share the A fragment);
//    main K loop is branch-free (addresses clamped in-range, OOB rows/cols
//    computed but never stored); guarded K-tail handled once.
// =====================================================================
#define GEMM_WAVES 4
template <int TRANSB>
__global__ __launch_bounds__(128) void gemm_wmma(
    const float* __restrict__ A, int sAy, int sAz, int lda,
    const float* __restrict__ B, int sBy, int sBz, int ldb,
    float* __restrict__ C, int sCy, int sCz, int ldc,
    const float* __restrict__ bias, int M, int N, int K, int relu)
{
  A += (size_t)blockIdx.y * sAy + (size_t)blockIdx.z * sAz;
  B += (size_t)blockIdx.y * sBy + (size_t)blockIdx.z * sBz;
  C += (size_t)blockIdx.y * sCy + (size_t)blockIdx.z * sCz;

  const int lane = threadIdx.x & 31;
  const int wave = threadIdx.x >> 5;
  const int tilesN = (N + 31) >> 5;
  const int tilesM = (M + 15) >> 4;
  const int tile = blockIdx.x * GEMM_WAVES + wave;
  if (tile >= tilesM * tilesN) return;            // wave-uniform exit
  const int tm = (tile / tilesN) << 4;
  const int tn = (tile % tilesN) << 5;

  const int l16 = lane & 15;
  const int hi  = lane >> 4;                      // 0: K lanes {0,1}, 1: {2,3}
  const int row  = tm + l16;                      // A row (M)
  const int col0 = tn + l16;                      // B col (N), first 16-tile
  const int col1 = col0 + 16;                     // second 16-tile
  // clamp addresses in-range; clamped lanes produce garbage that is not stored
  const int rowc  = (row  < M) ? row  : (M - 1);
  const int col0c = (col0 < N) ? col0 : (N - 1);
  const int col1c = (col1 < N) ? col1 : (N - 1);

  const float* pa = A + (size_t)rowc * lda + (hi << 1);
  const float* pb0;
  const float* pb1;
  if (TRANSB) {
    pb0 = B + (size_t)col0c * ldb + (hi << 1);
    pb1 = B + (size_t)col1c * ldb + (hi << 1);
  } else {
    pb0 = B + (size_t)(hi << 1) * ldb + col0c;
    pb1 = B + (size_t)(hi << 1) * ldb + col1c;
  }

  v8f acc0 = {}, acc1 = {};
  const int K4 = K & ~3;
  for (int k0 = 0; k0 < K4; k0 += 4) {
    v2f a, b0, b1;
    a.x = pa[0];
    a.y = pa[1];
    if (TRANSB) {
      b0.x = pb0[0]; b0.y = pb0[1];
      b1.x = pb1[0]; b1.y = pb1[1];
      pb0 += 4; pb1 += 4;
    } else {
      b0.x = pb0[0]; b0.y = pb0[ldb];
      b1.x = pb1[0]; b1.y = pb1[ldb];
      pb0 += (size_t)4 * ldb; pb1 += (size_t)4 * ldb;
    }
    pa += 4;
    acc0 = __builtin_amdgcn_wmma_f32_16x16x4_f32(false, a, false, b0,
                                                 (short)0, acc0, false, false);
    acc1 = __builtin_amdgcn_wmma_f32_16x16x4_f32(false, a, false, b1,
                                                 (short)0, acc1, false, false);
  }
  if (K4 < K) {                                   // K tail (1..3), once per tile
    const int ka = K4 + (hi << 1);
    const int kb = ka + 1;
    const bool ga = ka < K, gb = kb < K;
    v2f a, b0, b1;
    a.x = ga ? pa[0] : 0.0f;
    a.y = gb ? pa[1] : 0.0f;
    if (TRANSB) {
      b0.x = ga ? pb0[0] : 0.0f;  b0.y = gb ? pb0[1] : 0.0f;
      b1.x = ga ? pb1[0] : 0.0f;  b1.y = gb ? pb1[1] : 0.0f;
    } else {
      b0.x = ga ? pb0[0] : 0.0f;  b0.y = gb ? pb0[ldb] : 0.0f;
      b1.x = ga ? pb1[0] : 0.0f;  b1.y = gb ? pb1[ldb] : 0.0f;
    }
    acc0 = __builtin_amdgcn_wmma_f32_16x16x4_f32(false, a, false, b0,
                                                 (short)0, acc0, false, false);
    acc1 = __builtin_amdgcn_wmma_f32_16x16x4_f32(false, a, false, b1,
                                                 (short)0, acc1, false, false);
  }

  const float bv0 = (bias != nullptr && col0 < N) ? bias[col0] : 0.0f;
  const float bv1 = (bias != nullptr && col1 < N) ? bias[col1] : 0.0f;
  const int rbase = tm + (hi << 3);               // C/D layout: vgpr r -> row r (+8 hi)
#pragma unroll
  for (int r = 0; r < 8; ++r) {
    const int cr = rbase + r;
    if (cr < M) {
      if (col0 < N) {
        float v = acc0[r] + bv0;
        if (relu) v = fmaxf(v, 0.0f);
        C[(size_t)cr * ldc + col0] = v;
      }
      if (col1 < N) {
        float v = acc1[r] + bv1;
        if (relu) v = fmaxf(v, 0.0f);
        C[(size_t)cr * ldc + col1] = v;
      }
    }
  }
}

// =====================================================================
// Convs (small FLOP share -> VALU)
// =====================================================================
__global__ void conv_first_kernel(const float* __restrict__ x,
                                  const float* __restrict__ w,
                                  const float* __restrict__ bias,
                                  float* __restrict__ y)
{
  int idx = blockIdx.x * blockDim.x + threadIdx.x;
  const int total = CB * CC * CF * CL1;
  if (idx >= total) return;
  int l  = idx % CL1;
  int ff = (idx / CL1) % CF;
  int co = (idx / (CL1 * CF)) % CC;
  int b  = idx / (CL1 * CF * CC);
  float acc = bias[co];
  const float* wr = w + co * CC * 3;
  for (int ci = 0; ci < CC; ++ci) {
    const float* xp = x + (((size_t)(b * CC + ci) * CS + 2 * l) * CF + ff);
    acc += xp[0] * wr[ci * 3 + 0] + xp[CF] * wr[ci * 3 + 1] + xp[2 * CF] * wr[ci * 3 + 2];
  }
  y[idx] = acc;
}

__global__ void conv_rest_kernel(const float* __restrict__ in,
                                 const float* __restrict__ w,
                                 const float* __restrict__ bias,
                                 float* __restrict__ y, int Lin, int Lout)
{
  int idx = blockIdx.x * blockDim.x + threadIdx.x;
  const int total = CB * CC * CF * Lout;
  if (idx >= total) return;
  int l  = idx % Lout;
  int ff = (idx / Lout) % CF;
  int co = (idx / (Lout * CF)) % CC;
  int b  = idx / (Lout * CF * CC);
  float acc = bias[co];
  const float* wr = w + co * CC * 3;
  for (int ci = 0; ci < CC; ++ci) {
    const float* ip = in + ((size_t)(b * CC + ci) * CF + ff) * Lin + 2 * l;
    acc += ip[0] * wr[ci * 3 + 0] + ip[1] * wr[ci * 3 + 1] + ip[2] * wr[ci * 3 + 2];
  }
  y[idx] = acc;
}

// =====================================================================
// Positional embedding helpers
// =====================================================================
__global__ void build_pos1_kernel(const float* __restrict__ pe, float* __restrict__ p1)
{
  int idx = blockIdx.x * blockDim.x + threadIdx.x;
  if (idx >= CC * CL1) return;
  int l = idx % CL1, c = idx / CL1;
  p1[c * CL1 + l] = pe[l * CC + c];
}

__global__ void pool_pos_kernel(const float* __restrict__ pin, float* __restrict__ pout,
                                int Lin, int Lout)
{
  int idx = blockIdx.x * blockDim.x + threadIdx.x;
  if (idx >= CC * Lout) return;
  int i = idx % Lout, c = idx / Lout;
  int start = (i * Lin) / Lout;
  int end   = ((i + 1) * Lin + Lout - 1) / Lout;   // ceil
  float s = 0.0f;
  for (int l = start; l < end; ++l) s += pin[c * Lin + l];
  pout[c * Lout + i] = s / (float)(end - start);
}

__global__ void addpos_kernel(float* __restrict__ y, const float* __restrict__ posb, int L)
{
  int idx = blockIdx.x * blockDim.x + threadIdx.x;
  const int total = CB * CC * CF * L;
  if (idx >= total) return;
  int l = idx % L;
  int c = (idx / (L * CF)) % CC;
  y[idx] += posb[c * L + l];
}

// flat: f[b, l, ff*63+c] = y[b, c, ff, l]
__global__ void flat_kernel(const float* __restrict__ y, float* __restrict__ f, int L)
{
  int idx = blockIdx.x * blockDim.x + threadIdx.x;
  const int total = CB * L * CE;
  if (idx >= total) return;
  int j = idx % CE;
  int l = (idx / CE) % L;
  int b = idx / (CE * L);
  int c = j % CC, ff = j / CC;
  f[idx] = y[((size_t)(b * CC + c) * CF + ff) * (size_t)L + l];
}

// =====================================================================
// Row softmax (applies scale before exp)
// =====================================================================
__global__ void softmax_rows_kernel(float* __restrict__ S, int n, float scale)
{
  __shared__ float red[256];
  float* p = S + (size_t)blockIdx.x * n;
  int t = threadIdx.x;
  float m = -3.402823e38f;
  for (int i = t; i < n; i += 256) m = fmaxf(m, p[i] * scale);
  red[t] = m; __syncthreads();
  for (int s = 128; s > 0; s >>= 1) { if (t < s) red[t] = fmaxf(red[t], red[t + s]); __syncthreads(); }
  m = red[0]; __syncthreads();
  float sum = 0.0f;
  for (int i = t; i < n; i += 256) { float e = __expf(p[i] * scale - m); p[i] = e; sum += e; }
  red[t] = sum; __syncthreads();
  for (int s = 128; s > 0; s >>= 1) { if (t < s) red[t] += red[t + s]; __syncthreads(); }
  float inv = 1.0f / red[0];
  for (int i = t; i < n; i += 256) p[i] *= inv;
}

// =====================================================================
// GATv2 pieces (scatter via float atomics; max via monotone-uint encoding)
// =====================================================================
__device__ __forceinline__ unsigned fenc(float f) {
  unsigned u = __float_as_uint(f);
  return (u & 0x80000000u) ? ~u : (u | 0x80000000u);
}
__device__ __forceinline__ float fdec(unsigned e) {
  unsigned u = (e & 0x80000000u) ? (e ^ 0x80000000u) : ~e;
  return __uint_as_float(u);
}
#define FENC_NEG_INF 0x007FFFFFu   // fenc(-inf)

__global__ void fill_f32_kernel(float* p, int n, float v)
{ int i = blockIdx.x * blockDim.x + threadIdx.x; if (i < n) p[i] = v; }
__global__ void fill_u32_kernel(unsigned* p, int n, unsigned v)
{ int i = blockIdx.x * blockDim.x + threadIdx.x; if (i < n) p[i] = v; }

__global__ void gat_deg_kernel(const int* __restrict__ dst, const float* __restrict__ ea,
                               float* deg, float* lsum, int E)
{
  int e = blockIdx.x * blockDim.x + threadIdx.x;
  if (e >= E) return;
  int d = dst[e];
  atomicAdd(&deg[d], 1.0f);
  atomicAdd(&lsum[d], ea[e]);
}

__global__ void gat_loopattr_kernel(float* lsum, const float* deg, int N)
{
  int n = blockIdx.x * blockDim.x + threadIdx.x;
  if (n >= N) return;
  lsum[n] /= fmaxf(deg[n], 1.0f);
}

__global__ void gat_score_kernel(const float* __restrict__ xl, const float* __restrict__ xr,
                                 const int* __restrict__ src, const int* __restrict__ dst,
                                 const float* __restrict__ ea, const float* __restrict__ lattr,
                                 const float* __restrict__ we, const float* __restrict__ att,
                                 float* __restrict__ score, unsigned* __restrict__ smax,
                                 int E, int N)
{
  int e = blockIdx.x * blockDim.x + threadIdx.x;
  if (e >= E + N) return;
  int s, d; float a;
  if (e < E) { s = src[e]; d = dst[e]; a = ea[e]; }
  else       { s = d = e - E; a = lattr[s]; }
  const float* ps = xl + (size_t)s * CO;
  const float* pd = xr + (size_t)d * CO;
  float sc = 0.0f;
  for (int o = 0; o < CO; ++o) {
    float m = ps[o] + pd[o] + a * we[o];
    m = (m > 0.0f) ? m : 0.2f * m;              // leaky_relu(0.2)
    sc += m * att[o];
  }
  score[e] = sc;
  atomicMax(&smax[d], fenc(sc));
}

__global__ void gat_exp_kernel(const float* __restrict__ score, const unsigned* __restrict__ smax,
                               const int* __restrict__ dst, float* __restrict__ exs,
                               float* __restrict__ denom, int E, int N)
{
  int e = blockIdx.x * blockDim.x + threadIdx.x;
  if (e >= E + N) return;
  int d = (e < E) ? dst[e] : e - E;
  float ex = __expf(score[e] - fdec(smax[d]));
  exs[e] = ex;
  atomicAdd(&denom[d], ex);
}

__global__ void gat_accum_kernel(const float* __restrict__ xl, const float* __restrict__ exs,
                                 const float* __restrict__ denom, const int* __restrict__ src,
                                 const int* __restrict__ dst, float* __restrict__ outb,
                                 int E, int N)
{
  int e = blockIdx.x * blockDim.x + threadIdx.x;
  if (e >= E + N) return;
  int s, d;
  if (e < E) { s = src[e]; d = dst[e]; } else { s = d = e - E; }
  float alpha = exs[e] / denom[d];
  const float* ps = xl + (size_t)s * CO;
  float* pd = outb + (size_t)d * CO;
  for (int o = 0; o < CO; ++o) atomicAdd(&pd[o], alpha * ps[o]);
}

__global__ void gat_pool_kernel(const float* __restrict__ outb, const float* __restrict__ gbias,
                                float* __restrict__ pooled)
{
  int idx = blockIdx.x * blockDim.x + threadIdx.x;
  if (idx >= CG * CO) return;
  int o = idx & (CO - 1);
  int g = idx >> 5;
  const float* p = outb + (size_t)g * CC * CO + o;
  float s = 0.0f;
  for (int i = 0; i < CC; ++i) s += p[i * CO];
  pooled[idx] = s * (1.0f / (float)CC) + gbias[o];
}

// =====================================================================
// GRU scan: single workgroup (8 batches x 64 hidden = 512 threads),
// whh staged in LDS (192*64*4 = 48KB < 320KB/WGP)
// =====================================================================
__global__ __launch_bounds__(512) void gru_kernel(const float* __restrict__ xgates,
                                                  const float* __restrict__ whh,
                                                  const float* __restrict__ bhh,
                                                  float* __restrict__ out)
{
  __shared__ float sW[3 * CH * CH];   // [192][64]
  __shared__ float h[CB * CH];
  int t = threadIdx.x;
  for (int i = t; i < 3 * CH * CH; i += 512) sW[i] = whh[i];
  h[t] = 0.0f;
  __syncthreads();
  int b = t >> 6, j = t & 63;
  float br = bhh[j], bz = bhh[CH + j], bn = bhh[2 * CH + j];
  for (int step = 0; step < CL1; ++step) {
    float hr = br, hz = bz, hn = bn;
    const float* hb = h + b * CH;
    for (int k = 0; k < CH; ++k) {
      float hk = hb[k];
      hr += hk * sW[j * CH + k];
      hz += hk * sW[(CH + j) * CH + k];
      hn += hk * sW[(2 * CH + j) * CH + k];
    }
    const float* xg = xgates + (size_t)(b * CL1 + step) * (3 * CH);
    float r = 1.0f / (1.0f + __expf(-(xg[j] + hr)));
    float z = 1.0f / (1.0f + __expf(-(xg[CH + j] + hz)));
    float n = tanhf(xg[2 * CH + j] + r * hn);
    float hprev = h[b * CH + j];
    float hnew = (1.0f - z) * n + z * hprev;
    __syncthreads();
    h[b * CH + j] = hnew;
    out[(size_t)(b * CL1 + step) * CH + j] = hnew;
    __syncthreads();
  }
}

// =====================================================================
// top-k (K=5) over L1 per (batch, class); stable ties like jax top_k
// =====================================================================
__global__ void topk_kernel(const float* __restrict__ scores, float* __restrict__ dval,
                            int* __restrict__ didx)
{
  int t = threadIdx.x;
  if (t >= CB * 3) return;
  int b = t / 3, c = t % 3;
  float v[5]; int id[5];
  for (int i = 0; i < 5; ++i) { v[i] = -3.402823e38f; id[i] = 0; }
  for (int s = 0; s < CL1; ++s) {
    float x = scores[(size_t)(b * CL1 + s) * 3 + c];
    if (x > v[4]) {
      int p = 4;
      while (p > 0 && x > v[p - 1]) { v[p] = v[p - 1]; id[p] = id[p - 1]; --p; }
      v[p] = x; id[p] = s;
    }
  }
  dval[b * 3 + c] = (v[0] + v[1] + v[2] + v[3] + v[4]) * 0.2f;
  for (int k = 0; k < 5; ++k) didx[(b * 5 + k) * 3 + c] = id[k];
}

// =====================================================================
// host-side launch
// =====================================================================
static inline dim3 g1(int n) { return dim3((n + 255) / 256); }

static void gemm(hipStream_t st,
                 const float* A, int sAy, int sAz, int lda,
                 const float* B, int sBy, int sBz, int ldb, int transB,
                 float* C, int sCy, int sCz, int ldc,
                 const float* bias, int M, int N, int K, int relu, int gy, int gz)
{
  int tiles = ((M + 15) / 16) * ((N + 31) / 32);
  dim3 grid((tiles + GEMM_WAVES - 1) / GEMM_WAVES, gy, gz);
  if (transB)
    gemm_wmma<1><<<grid, dim3(128), 0, st>>>(A, sAy, sAz, lda, B, sBy, sBz, ldb,
                                             C, sCy, sCz, ldc, bias, M, N, K, relu);
  else
    gemm_wmma<0><<<grid, dim3(128), 0, st>>>(A, sAy, sAz, lda, B, sBy, sBz, ldb,
                                             C, sCy, sCz, ldc, bias, M, N, K, relu);
}

extern "C" void kernel_launch(void* const* d_in, const int* in_sizes, int n_in,
                              void* d_out, int out_size, void* d_ws, size_t ws_size,
                              hipStream_t stream)
{
  const float* x    = (const float*)d_in[0];
  const float* c1w  = (const float*)d_in[1];  const float* c1b = (const float*)d_in[2];
  const float* c2w  = (const float*)d_in[3];  const float* c2b = (const float*)d_in[4];
  const float* c3w  = (const float*)d_in[5];  const float* c3b = (const float*)d_in[6];
  const float* pe   = (const float*)d_in[7];
  const float* aw   = (const float*)d_in[8];  const float* ab  = (const float*)d_in[9];
  const float* aow  = (const float*)d_in[10]; const float* aob = (const float*)d_in[11];
  const float* gwl  = (const float*)d_in[12]; const float* gbl = (const float*)d_in[13];
  const float* gwr  = (const float*)d_in[14]; const float* gbr = (const float*)d_in[15];
  const float* gwe  = (const float*)d_in[16]; const float* gatt= (const float*)d_in[17];
  const float* gbias= (const float*)d_in[18];
  const float* wih  = (const float*)d_in[19]; const float* whh = (const float*)d_in[20];
  const float* bih  = (const float*)d_in[21]; const float* bhh = (const float*)d_in[22];
  const float* fw1  = (const float*)d_in[23]; const float* fb1 = (const float*)d_in[24];
  const float* fw2  = (const float*)d_in[25]; const float* fb2 = (const float*)d_in[26];
  const float* ea   = (const float*)d_in[27];
  const int*   eidx = (const int*)d_in[28];
  const int E = in_sizes[27];                 // edge_attr count = #edges
  const int* esrc = eidx;
  const int* edst = eidx + E;

  // bump allocator over d_ws
  size_t off = 0;
  auto alloc = [&](size_t n) -> float* {
    float* p = (float*)((char*)d_ws + off);
    off += ((n * sizeof(float)) + 255) & ~(size_t)255;
    return p;
  };

  float* y1   = alloc((size_t)CB * CC * CF * CL1);
  float* y2   = alloc((size_t)CB * CC * CF * CL2);
  float* y3   = alloc((size_t)CB * CC * CF * CL3);
  float* pos1 = alloc((size_t)CC * CL1);
  float* pos2 = alloc((size_t)CC * CL2);
  float* pos3 = alloc((size_t)CC * CL3);
  float* f1   = alloc((size_t)CB * CL1 * CE);
  float* f2   = alloc((size_t)CB * CL2 * CE);
  float* f3   = alloc((size_t)CB * CL3 * CE);
  float* Qp1  = alloc((size_t)CB * CL2 * CE);
  float* Kp1  = alloc((size_t)CB * CL3 * CE);
  float* Vp1  = alloc((size_t)CB * CL3 * CE);
  float* S1   = alloc((size_t)CB * 3 * CL2 * CL3);
  float* O1   = alloc((size_t)CB * CL2 * CE);
  float* x2a  = alloc((size_t)CB * CL2 * CE);
  float* Qp2  = alloc((size_t)CB * CL1 * CE);
  float* Kp2  = alloc((size_t)CB * CL2 * CE);
  float* Vp2  = alloc((size_t)CB * CL2 * CE);
  float* S2   = alloc((size_t)CB * 3 * CL1 * CL2);
  float* O2   = alloc((size_t)CB * CL1 * CE);
  float* x1a  = alloc((size_t)CB * CL1 * CE);
  float* xl   = alloc((size_t)CN * CO);
  float* xr   = alloc((size_t)CN * CO);
  float* deg  = alloc((size_t)CN);
  float* lsum = alloc((size_t)CN);
  float* score= alloc((size_t)(E + CN));
  float* exs  = alloc((size_t)(E + CN));
  unsigned* smax = (unsigned*)alloc((size_t)CN);
  float* denom= alloc((size_t)CN);
  float* gout = alloc((size_t)CN * CO);
  float* pooled = alloc((size_t)CG * CO);
  float* xgates = alloc((size_t)CG * 3 * CH);
  float* gruo   = alloc((size_t)CG * CH);
  float* s1buf  = alloc((size_t)CG * (CH / 2));
  float* sc3    = alloc((size_t)CG * 3);
  (void)ws_size; (void)n_in; (void)out_size;

  // ---- Encoder convs ----
  conv_first_kernel<<<g1(CB*CC*CF*CL1), 256, 0, stream>>>(x, c1w, c1b, y1);
  conv_rest_kernel<<<g1(CB*CC*CF*CL2), 256, 0, stream>>>(y1, c2w, c2b, y2, CL1, CL2);
  conv_rest_kernel<<<g1(CB*CC*CF*CL3), 256, 0, stream>>>(y2, c3w, c3b, y3, CL2, CL3);

  // ---- positional embeddings (added after conv chain) ----
  build_pos1_kernel<<<g1(CC*CL1), 256, 0, stream>>>(pe, pos1);
  pool_pos_kernel<<<g1(CC*CL2), 256, 0, stream>>>(pos1, pos2, CL1, CL2);
  pool_pos_kernel<<<g1(CC*CL3), 256, 0, stream>>>(pos2, pos3, CL2, CL3);
  addpos_kernel<<<g1(CB*CC*CF*CL1), 256, 0, stream>>>(y1, pos1, CL1);
  addpos_kernel<<<g1(CB*CC*CF*CL2), 256, 0, stream>>>(y2, pos2, CL2);
  addpos_kernel<<<g1(CB*CC*CF*CL3), 256, 0, stream>>>(y3, pos3, CL3);

  // ---- flatten to [B, L, 315] ----
  flat_kernel<<<g1(CB*CL1*CE), 256, 0, stream>>>(y1, f1, CL1);
  flat_kernel<<<g1(CB*CL2*CE), 256, 0, stream>>>(y2, f2, CL2);
  flat_kernel<<<g1(CB*CL3*CE), 256, 0, stream>>>(y3, f3, CL3);

  const float attn_scale = 1.0f / sqrtf((float)CHD);
  const float* wq = aw;                const float* bq = ab;
  const float* wk = aw + CE * CE;      const float* bk = ab + CE;
  const float* wv = aw + 2 * CE * CE;  const float* bv = ab + 2 * CE;

  // ---- MHA #1: q = f2 [B,511,E], k=v = f3 [B,255,E] ----
  gemm(stream, f2,0,0,CE,  wq,0,0,CE,1,  Qp1,0,0,CE, bq, CB*CL2, CE, CE, 0, 1,1);
  gemm(stream, f3,0,0,CE,  wk,0,0,CE,1,  Kp1,0,0,CE, bk, CB*CL3, CE, CE, 0, 1,1);
  gemm(stream, f3,0,0,CE,  wv,0,0,CE,1,  Vp1,0,0,CE, bv, CB*CL3, CE, CE, 0, 1,1);
  // scores[b,h] = Qp1[b,:,h*105:] @ Kp1[b,:,h*105:]^T
  gemm(stream, Qp1,CHD,CL2*CE,CE,  Kp1,CHD,CL3*CE,CE,1,
       S1,CL2*CL3,3*CL2*CL3,CL3, nullptr, CL2, CL3, CHD, 0, 3, CB);
  softmax_rows_kernel<<<dim3(CB*3*CL2), 256, 0, stream>>>(S1, CL3, attn_scale);
  gemm(stream, S1,CL2*CL3,3*CL2*CL3,CL3,  Vp1,CHD,CL3*CE,CE,0,
       O1,CHD,CL2*CE,CE, nullptr, CL2, CHD, CL3, 0, 3, CB);
  gemm(stream, O1,0,0,CE,  aow,0,0,CE,1,  x2a,0,0,CE, aob, CB*CL2, CE, CE, 0, 1,1);

  // ---- MHA #2: q = f1 [B,1023,E], k=v = x2a [B,511,E] ----
  gemm(stream, f1,0,0,CE,   wq,0,0,CE,1,  Qp2,0,0,CE, bq, CB*CL1, CE, CE, 0, 1,1);
  gemm(stream, x2a,0,0,CE,  wk,0,0,CE,1,  Kp2,0,0,CE, bk, CB*CL2, CE, CE, 0, 1,1);
  gemm(stream, x2a,0,0,CE,  wv,0,0,CE,1,  Vp2,0,0,CE, bv, CB*CL2, CE, CE, 0, 1,1);
  gemm(stream, Qp2,CHD,CL1*CE,CE,  Kp2,CHD,CL2*CE,CE,1,
       S2,CL1*CL2,3*CL1*CL2,CL2, nullptr, CL1, CL2, CHD, 0, 3, CB);
  softmax_rows_kernel<<<dim3(CB*3*CL1), 256, 0, stream>>>(S2, CL2, attn_scale);
  gemm(stream, S2,CL1*CL2,3*CL1*CL2,CL2,  Vp2,CHD,CL2*CE,CE,0,
       O2,CHD,CL1*CE,CE, nullptr, CL1, CHD, CL2, 0, 3, CB);
  gemm(stream, O2,0,0,CE,  aow,0,0,CE,1,  x1a,0,0,CE, aob, CB*CL1, CE, CE, 0, 1,1);

  // ---- GATv2: xg = x1a viewed as [N, 5] ----
  gemm(stream, x1a,0,0,CF, gwl,0,0,CF,1, xl,0,0,CO, gbl, CN, CO, CF, 0, 1,1);
  gemm(stream, x1a,0,0,CF, gwr,0,0,CF,1, xr,0,0,CO, gbr, CN, CO, CF, 0, 1,1);
  fill_f32_kernel<<<g1(CN), 256, 0, stream>>>(deg, CN, 0.0f);
  fill_f32_kernel<<<g1(CN), 256, 0, stream>>>(lsum, CN, 0.0f);
  fill_f32_kernel<<<g1(CN), 256, 0, stream>>>(denom, CN, 0.0f);
  fill_u32_kernel<<<g1(CN), 256, 0, stream>>>(smax, CN, FENC_NEG_INF);
  fill_f32_kernel<<<g1(CN*CO), 256, 0, stream>>>(gout, CN*CO, 0.0f);
  gat_deg_kernel<<<g1(E), 256, 0, stream>>>(edst, ea, deg, lsum, E);
  gat_loopattr_kernel<<<g1(CN), 256, 0, stream>>>(lsum, deg, CN);
  gat_score_kernel<<<g1(E+CN), 256, 0, stream>>>(xl, xr, esrc, edst, ea, lsum,
                                                 gwe, gatt, score, smax, E, CN);
  gat_exp_kernel<<<g1(E+CN), 256, 0, stream>>>(score, smax, edst, exs, denom, E, CN);
  gat_accum_kernel<<<g1(E+CN), 256, 0, stream>>>(xl, exs, denom, esrc, edst, gout, E, CN);
  gat_pool_kernel<<<g1(CG*CO), 256, 0, stream>>>(gout, gbias, pooled);

  // ---- GRU ----
  gemm(stream, pooled,0,0,CO, wih,0,0,CO,1, xgates,0,0,3*CH, bih, CG, 3*CH, CO, 0, 1,1);
  gru_kernel<<<dim3(1), dim3(512), 0, stream>>>(xgates, whh, bhh, gruo);

  // ---- classifier + top-k ----
  gemm(stream, gruo,0,0,CH,  fw1,0,0,CH,1, s1buf,0,0,CH/2, fb1, CG, CH/2, CH, 1, 1,1);
  gemm(stream, s1buf,0,0,CH/2, fw2,0,0,CH/2,1, sc3,0,0,3,  fb2, CG, 3, CH/2, 0, 1,1);
  topk_kernel<<<dim3(1), dim3(32), 0, stream>>>(sc3, (float*)d_out, (int*)d_out + CB*3);
}